// MoELayer_25220047962117
// MI455X (gfx1250) — compile-verified
//
#include <hip/hip_runtime.h>
#include <math.h>

#define T_TOK 8192
#define H_DIM 1024
#define F_DIM 4096
#define E_NUM 8

#define CHUNK_F 256
#define LDX 1044   // X-tile leading dim: 16B-aligned rows for async b128 writes,
                   // 20-dword bank shift/row -> conflict-free f32x2 A-frag reads
#define LDA 260    // activation chunk leading dim (4-dword shift/row, conflict-free)

typedef __attribute__((ext_vector_type(2))) float v2f;
typedef __attribute__((ext_vector_type(8))) float v8f;

__device__ __forceinline__ float gelu_exact(float z) {
    return 0.5f * z * (1.0f + erff(z * 0.70710678118654752f));
}

// Byte offset of a (generic) pointer into LDS, for async-to-LDS addressing.
__device__ __forceinline__ unsigned lds_offset(const void* p) {
    return (unsigned)(uintptr_t)(__attribute__((address_space(3))) const void*)p;
}

// ---------------------------------------------------------------------------
// Router: logits = x@Wr + br, softmax, top-2 gates, dispatch; compact
// (token, gate) pairs per expert into workspace; accumulate aux-loss sums.
// One wave (32 lanes) per token.
// ---------------------------------------------------------------------------
__global__ void router_kernel(const float* __restrict__ x,
                              const float* __restrict__ Wr,
                              const float* __restrict__ br,
                              float* __restrict__ rw_out,
                              float* __restrict__ disp_out,
                              float* __restrict__ logits_out,
                              int*   __restrict__ counts,
                              float* __restrict__ sum_rw,
                              float* __restrict__ sum_disp,
                              int2*  __restrict__ pairs)
{
    const int lane = threadIdx.x & 31;
    const int wv   = threadIdx.x >> 5;
    const int t    = blockIdx.x * 8 + wv;
    if (t >= T_TOK) return;

    float acc[E_NUM];
#pragma unroll
    for (int e = 0; e < E_NUM; ++e) acc[e] = 0.f;

    const float4* wr4 = reinterpret_cast<const float4*>(Wr);
    const float*  xr  = x + (size_t)t * H_DIM;
    for (int h = lane; h < H_DIM; h += 32) {
        float  xv = xr[h];
        float4 wa = wr4[h * 2 + 0];
        float4 wb = wr4[h * 2 + 1];
        acc[0] += xv * wa.x; acc[1] += xv * wa.y;
        acc[2] += xv * wa.z; acc[3] += xv * wa.w;
        acc[4] += xv * wb.x; acc[5] += xv * wb.y;
        acc[6] += xv * wb.z; acc[7] += xv * wb.w;
    }
#pragma unroll
    for (int e = 0; e < E_NUM; ++e) {
#pragma unroll
        for (int off = 16; off >= 1; off >>= 1)
            acc[e] += __shfl_xor(acc[e], off, 32);
        acc[e] += br[e];
    }

    // full softmax (routing_weights)
    float m = acc[0];
#pragma unroll
    for (int e = 1; e < E_NUM; ++e) m = fmaxf(m, acc[e]);
    float s = 0.f, rw[E_NUM];
#pragma unroll
    for (int e = 0; e < E_NUM; ++e) { rw[e] = expf(acc[e] - m); s += rw[e]; }
    const float inv = 1.f / s;

    // top-2 (stable: lowest index wins ties, matching lax.top_k)
    int i0 = 0;
#pragma unroll
    for (int e = 1; e < E_NUM; ++e) if (acc[e] > acc[i0]) i0 = e;
    int i1 = (i0 == 0) ? 1 : 0;
#pragma unroll
    for (int e = 0; e < E_NUM; ++e) if (e != i0 && acc[e] > acc[i1]) i1 = e;
    const float r10 = expf(acc[i1] - acc[i0]);   // <= 1
    const float g0  = 1.f / (1.f + r10);
    const float g1  = r10 * g0;

    if (lane < E_NUM) {
        const int e = lane;
        const float rwe = rw[e] * inv;
        logits_out[(size_t)t * E_NUM + e] = acc[e];
        rw_out[(size_t)t * E_NUM + e]     = rwe;
        const float d = (e == i0) ? g0 : ((e == i1) ? g1 : 0.f);
        disp_out[(size_t)t * E_NUM + e]   = d;
        atomicAdd(&sum_rw[e], rwe);
        if (d != 0.f) atomicAdd(&sum_disp[e], d);
    }
    if (lane == 0) {
        int p0 = atomicAdd(&counts[i0], 1);
        pairs[(size_t)i0 * T_TOK + p0] = make_int2(t, __float_as_int(g0));
        int p1 = atomicAdd(&counts[i1], 1);
        pairs[(size_t)i1 * T_TOK + p1] = make_int2(t, __float_as_int(g1));
    }
}

// ---------------------------------------------------------------------------
// aux_loss = E * sum_e (mean_t rw[t,e]) * (sum_t disp[t,e] / T)
// ---------------------------------------------------------------------------
__global__ void aux_kernel(const float* __restrict__ sum_rw,
                           const float* __restrict__ sum_disp,
                           float* __restrict__ aux_out)
{
    if (threadIdx.x == 0 && blockIdx.x == 0) {
        float s = 0.f;
#pragma unroll
        for (int e = 0; e < E_NUM; ++e) s += sum_rw[e] * sum_disp[e];
        *aux_out = s * (float)E_NUM / ((float)T_TOK * (float)T_TOK);
    }
}

// ---------------------------------------------------------------------------
// Expert FFN over compacted token lists, fp32 WMMA (V_WMMA_F32_16X16X4_F32).
// X tile gathered memory->LDS via GLOBAL_LOAD_ASYNC_TO_LDS_B128 (ASYNCcnt).
// Block: 256 threads (8 waves). Each block: one expert, one 16-token tile.
//   GEMM1: Z[16x256 chunk] = X[16x1024] @ W1[:, chunk]   (per wave: 2 N-tiles)
//   a = gelu(g*z + b1) -> LDS
//   GEMM2: C[16x1024] += a[16x256] @ W2[chunk, :]        (per wave: 8 N-tiles)
// Epilogue: out[token] += g * (C + b2)  via f32 atomics (2 experts/token).
// ---------------------------------------------------------------------------
__global__ __launch_bounds__(256) void ffn_kernel(
    const float* __restrict__ x,
    const float* __restrict__ W1,
    const float* __restrict__ b1,
    const float* __restrict__ W2,
    const float* __restrict__ b2,
    const int*   __restrict__ counts,
    const int2*  __restrict__ pairs,
    float* __restrict__ out)
{
    const int e     = blockIdx.y;
    const int cnt   = counts[e];
    const int tile0 = blockIdx.x * 16;
    if (tile0 >= cnt) return;        // over-launched tile: nothing routed here

    extern __shared__ float smem[];
    float* Xs    = smem;                    // [16][LDX]  raw x rows (gathered)
    float* As    = smem + 16 * LDX;         // [16][LDA]  gelu activations chunk
    float* gateS = As + 16 * LDA;           // [16]
    int*   tokS  = (int*)(gateS + 16);      // [16]

    const int tid = threadIdx.x;
    if (tid < 16) {
        const int pi = tile0 + tid;
        int2 pr = (pi < cnt) ? pairs[(size_t)e * T_TOK + pi] : make_int2(0, 0);
        tokS[tid]  = pr.x;
        gateS[tid] = (pi < cnt) ? __int_as_float(pr.y) : 0.f;  // pad rows: g=0
    }
    __syncthreads();

    // Async gather of 16 token rows (16 x 4KB) straight into LDS.
    // 16 x 256 float4 chunks over 256 threads -> 16 b128 asyncs per lane.
    {
        const unsigned xsBase = lds_offset(Xs);
        for (int idx = tid; idx < 16 * (H_DIM / 4); idx += 256) {
            const int r  = idx >> 8;        // token row 0..15
            const int c4 = idx & 255;       // float4 chunk within row
            const float* gp = x + (size_t)tokS[r] * H_DIM + (size_t)c4 * 4;
            const unsigned laddr = xsBase + (unsigned)((r * LDX + c4 * 4) * 4);
            asm volatile("global_load_async_to_lds_b128 %0, %1, off"
                         :: "v"(laddr), "v"((unsigned long long)(uintptr_t)gp)
                         : "memory");
        }
        asm volatile("s_wait_asynccnt 0x0" ::: "memory");
    }
    __syncthreads();

    const int lane  = tid & 31;
    const int wv    = tid >> 5;      // 0..7
    const int hi    = lane >> 4;     // 0/1 -> K/M split per WMMA layout
    const int lo    = lane & 15;
    const int klane = hi * 2;

    float gRow[8];
#pragma unroll
    for (int r = 0; r < 8; ++r) gRow[r] = gateS[r + 8 * hi];

    // GEMM2 accumulators: cols [wv*128, wv*128+128) as 8 16x16 tiles.
    v8f acc2[8];
#pragma unroll
    for (int ti = 0; ti < 8; ++ti) acc2[ti] = (v8f){0.f,0.f,0.f,0.f,0.f,0.f,0.f,0.f};

    const float* W1e = W1 + (size_t)e * H_DIM * F_DIM;
    const float* W2e = W2 + (size_t)e * F_DIM * H_DIM;

    for (int ch = 0; ch < F_DIM / CHUNK_F; ++ch) {
        const int fBase = ch * CHUNK_F;

        // ---- GEMM1: two 16x16 output tiles per wave ----
        v8f z0 = (v8f){0.f,0.f,0.f,0.f,0.f,0.f,0.f,0.f};
        v8f z1 = (v8f){0.f,0.f,0.f,0.f,0.f,0.f,0.f,0.f};
        const int col0 = fBase + wv * 32 + lo;
        const float* bp = W1e + (size_t)klane * F_DIM + col0;
        const float* ap = Xs + lo * LDX + klane;
        for (int k = 0; k < H_DIM; k += 4) {
            float2 af = *reinterpret_cast<const float2*>(ap);   // ds_load_b64
            v2f a; a.x = af.x; a.y = af.y;
            v2f b0; b0.x = bp[0];  b0.y = bp[F_DIM];
            v2f b1v; b1v.x = bp[16]; b1v.y = bp[16 + F_DIM];
            z0 = __builtin_amdgcn_wmma_f32_16x16x4_f32(false, a, false, b0,  (short)0, z0, false, false);
            z1 = __builtin_amdgcn_wmma_f32_16x16x4_f32(false, a, false, b1v, (short)0, z1, false, false);
            ap += 4;
            bp += (size_t)4 * F_DIM;
        }

        __syncthreads();   // previous chunk's GEMM2 must be done reading As

        // epilogue: a = gelu(g*z + b1) -> LDS
        {
            const float bias0 = b1[(size_t)e * F_DIM + col0];
            const float bias1 = b1[(size_t)e * F_DIM + col0 + 16];
            const int lc0 = col0 - fBase;
#pragma unroll
            for (int r = 0; r < 8; ++r) {
                const int mrow = r + 8 * hi;
                As[mrow * LDA + lc0]      = gelu_exact(gRow[r] * z0[r] + bias0);
                As[mrow * LDA + lc0 + 16] = gelu_exact(gRow[r] * z1[r] + bias1);
            }
        }
        __syncthreads();

        // ---- GEMM2: accumulate 8 N-tiles per wave over this K-chunk ----
        const int colBase2 = wv * 128 + lo;
        const float* a2p = As + lo * LDA + klane;
        const float* wRow = W2e + (size_t)(fBase + klane) * H_DIM + colBase2;
        for (int k = 0; k < CHUNK_F; k += 4) {
            float2 af = *reinterpret_cast<const float2*>(a2p);  // ds_load_b64
            v2f a; a.x = af.x; a.y = af.y;
#pragma unroll
            for (int ti = 0; ti < 8; ++ti) {
                v2f b; b.x = wRow[ti * 16]; b.y = wRow[ti * 16 + H_DIM];
                acc2[ti] = __builtin_amdgcn_wmma_f32_16x16x4_f32(false, a, false, b, (short)0, acc2[ti], false, false);
            }
            a2p += 4;
            wRow += (size_t)4 * H_DIM;
        }
    }

    // Final epilogue: out[token] += g * (acc + b2)
#pragma unroll
    for (int ti = 0; ti < 8; ++ti) {
        const int col = wv * 128 + ti * 16 + lo;
        const float bias = b2[(size_t)e * H_DIM + col];
#pragma unroll
        for (int r = 0; r < 8; ++r) {
            const int mrow = r + 8 * hi;
            const float val = gRow[r] * (acc2[ti][r] + bias);
            atomicAdd(&out[(size_t)tokS[mrow] * H_DIM + col], val);
        }
    }
}

// ---------------------------------------------------------------------------
extern "C" void kernel_launch(void* const* d_in, const int* in_sizes, int n_in,
                              void* d_out, int out_size, void* d_ws, size_t ws_size,
                              hipStream_t stream)
{
    (void)in_sizes; (void)n_in; (void)out_size; (void)ws_size;

    const float* x  = (const float*)d_in[0];
    const float* Wr = (const float*)d_in[1];
    const float* br = (const float*)d_in[2];
    const float* W1 = (const float*)d_in[3];
    const float* b1 = (const float*)d_in[4];
    const float* W2 = (const float*)d_in[5];
    const float* b2 = (const float*)d_in[6];

    float* out        = (float*)d_out;
    float* final_out  = out;                                   // [T, H]
    float* rw_out     = final_out + (size_t)T_TOK * H_DIM;     // [T, E]
    float* disp_out   = rw_out + (size_t)T_TOK * E_NUM;        // [T, E]
    float* aux_out    = disp_out + (size_t)T_TOK * E_NUM;      // [1]
    float* logits_out = aux_out + 1;                           // [T, E]

    char*  ws       = (char*)d_ws;
    int*   counts   = (int*)ws;                 // [E]
    float* sum_rw   = (float*)(ws + 64);        // [E]
    float* sum_disp = (float*)(ws + 96);        // [E]
    int2*  pairs    = (int2*)(ws + 256);        // [E][T] (token, gate-bits)

    hipMemsetAsync(d_ws, 0, 256, stream);
    hipMemsetAsync(final_out, 0, (size_t)T_TOK * H_DIM * sizeof(float), stream);

    router_kernel<<<T_TOK / 8, 256, 0, stream>>>(
        x, Wr, br, rw_out, disp_out, logits_out, counts, sum_rw, sum_disp, pairs);

    aux_kernel<<<1, 32, 0, stream>>>(sum_rw, sum_disp, aux_out);

    const size_t smem = (size_t)(16 * LDX + 16 * LDA + 16) * sizeof(float)
                      + (size_t)16 * sizeof(int);
    dim3 grid(T_TOK / 16, E_NUM);
    ffn_kernel<<<grid, 256, smem, stream>>>(x, W1, b1, W2, b2, counts, pairs, final_out);
}